// DiffTreeMachine_58669253263508
// MI455X (gfx1250) — compile-verified
//
#include <hip/hip_runtime.h>
#include <math.h>

typedef __attribute__((ext_vector_type(2))) float v2f;
typedef __attribute__((ext_vector_type(8))) float v8f;

#define Bn 64
#define Ln 16
#define Fn 64
#define Rn 1023
#define ROWS   (Bn * Fn)        // 4096
#define PLANE  (Fn * Rn)        // 65472
#define ARGSZ  (Bn * Fn * Rn)   // 4190208

// ---------------------------------------------------------------------------
// Stage 1: fused weighted sums over L for all four weight vectors.
// Reads x exactly once (268 MB), writes 4 x (B,F,R) arrays into workspace.
// ---------------------------------------------------------------------------
__global__ void __launch_bounds__(256)
dtm_stage1(const float* __restrict__ x,
           const float* __restrict__ w0, const float* __restrict__ w1,
           const float* __restrict__ w2, const float* __restrict__ w3,
           float* __restrict__ ws) {
  int idx = blockIdx.x * blockDim.x + threadIdx.x;
  if (idx >= ARGSZ) return;
  int b  = idx / PLANE;
  int fr = idx - b * PLANE;
  const float* xb = x + (size_t)b * Ln * PLANE + fr;
  float a0 = 0.f, a1 = 0.f, a2 = 0.f, a3 = 0.f;
#pragma unroll
  for (int l = 0; l < Ln; ++l) {
    float v = xb[(size_t)l * PLANE];
    a0 += v * w0[b * Ln + l];
    a1 += v * w1[b * Ln + l];
    a2 += v * w2[b * Ln + l];
    a3 += v * w3[b * Ln + l];
  }
  ws[idx]             = a0;   // car_arg
  ws[ARGSZ + idx]     = a1;   // cdr_arg
  ws[2 * ARGSZ + idx] = a2;   // c1
  ws[3 * ARGSZ + idx] = a3;   // c2
}

// ---------------------------------------------------------------------------
// Entropy + max per batch row for the four weight vectors (tiny).
// ---------------------------------------------------------------------------
__global__ void dtm_stats(const float* __restrict__ w0, const float* __restrict__ w1,
                          const float* __restrict__ w2, const float* __restrict__ w3,
                          float* __restrict__ out) {
  int tid = threadIdx.x;          // 256 threads: 4 weight arrays x 64 batches
  int wi  = tid >> 6;
  int b   = tid & 63;
  const float* w = (wi == 0) ? w0 : (wi == 1) ? w1 : (wi == 2) ? w2 : w3;
  float ent = 0.f, mx = -3.402823466e38f;
#pragma unroll
  for (int l = 0; l < Ln; ++l) {
    float p = w[b * Ln + l];
    ent -= p * logf(p + 1e-12f);
    mx   = fmaxf(mx, p);
  }
  ent *= (1.0f / 2.7725887222397811f);  // / log(16)
  out[(size_t)3 * ARGSZ + wi * Bn + b]           = ent;  // 4 entropies
  out[(size_t)3 * ARGSZ + 4 * Bn + wi * Bn + b]  = mx;   // 4 maxes
}

// ---------------------------------------------------------------------------
// Stage 2: fp32 WMMA GEMM path, fully specialized per output so pass count
// and matrix pointers are compile-time constants (scalar control flow only).
//   out[row, t] = sum_r arg[row, r] * M[t, r]   (+ rank-1 term for cons)
// One wave computes a 16-row x 64-col strip: 4 C tiles, A fragment reused 4x.
// ---------------------------------------------------------------------------
template <int OUT_ID>
__device__ __forceinline__ void gemm_path(
    int wid,                     // 0..4095 within this output's wave group
    int lane,
    const float* __restrict__ argA,   // (4096, 1023); cons: c1, c2 follows at +ARGSZ
    const float* __restrict__ M0,     // (1023, 1023) row-major
    const float* __restrict__ M1,     // second matrix (cons only)
    const float* __restrict__ root_filler,
    const float* __restrict__ root_role,
    float* __restrict__ dst) {        // (4096, 1023)
  int lo = lane & 15;           // A: row-in-tile; B & C/D: col-in-tile
  int hi = lane >> 4;           // K half-select for A/B; row half for C/D

  int row_tile = wid >> 4;      // 256 row tiles
  int colg     = wid & 15;      // 16 column groups of 64
  int rowbase  = row_tile * 16;
  int tbase    = colg * 64;

  v8f acc0 = (v8f)0.0f, acc1 = (v8f)0.0f, acc2 = (v8f)0.0f, acc3 = (v8f)0.0f;

  // Clamp column index so the never-stored t==1023 lane stays in-bounds.
  int t0 = tbase + lo;
  int t1 = t0 + 16;
  int t2 = t0 + 32;
  int t3c = t0 + 48; if (t3c > Rn - 1) t3c = Rn - 1;

  const int kOff = 2 * hi;      // lanes 16-31 carry K+2, K+3
  constexpr int NPASS = (OUT_ID == 2) ? 2 : 1;

#pragma unroll
  for (int pass = 0; pass < NPASS; ++pass) {
    const float* Mm = (pass == 0) ? M0 : M1;
    const float* aR = argA + (size_t)pass * ARGSZ + (size_t)(rowbase + lo) * Rn + kOff;
    const float* b0 = Mm + (size_t)t0 * Rn + kOff;
    const float* b1 = Mm + (size_t)t1 * Rn + kOff;
    const float* b2 = Mm + (size_t)t2 * Rn + kOff;
    const float* b3 = Mm + (size_t)t3c * Rn + kOff;

    // 255 full K=4 steps cover r = 0..1019 ; 255 = 3 * 85 (clean unroll)
#pragma unroll 3
    for (int step = 0; step < 255; ++step) {
      int kb = step * 4;
      v2f a, f0, f1, f2, f3;
      a.x  = aR[kb];     a.y  = aR[kb + 1];
      f0.x = b0[kb];     f0.y = b0[kb + 1];
      f1.x = b1[kb];     f1.y = b1[kb + 1];
      f2.x = b2[kb];     f2.y = b2[kb + 1];
      f3.x = b3[kb];     f3.y = b3[kb + 1];
      acc0 = __builtin_amdgcn_wmma_f32_16x16x4_f32(false, a, false, f0, (short)0, acc0, false, false);
      acc1 = __builtin_amdgcn_wmma_f32_16x16x4_f32(false, a, false, f1, (short)0, acc1, false, false);
      acc2 = __builtin_amdgcn_wmma_f32_16x16x4_f32(false, a, false, f2, (short)0, acc2, false, false);
      acc3 = __builtin_amdgcn_wmma_f32_16x16x4_f32(false, a, false, f3, (short)0, acc3, false, false);
    }
    // tail: r = 1020..1022 valid, r = 1023 zero-padded (hi lanes' 2nd elem)
    {
      int kb = 1020;
      bool ok = (kb + kOff + 1) < Rn;
      v2f a, f0, f1, f2, f3;
      a.x  = aR[kb];     a.y  = ok ? aR[kb + 1] : 0.f;
      f0.x = b0[kb];     f0.y = ok ? b0[kb + 1] : 0.f;
      f1.x = b1[kb];     f1.y = ok ? b1[kb + 1] : 0.f;
      f2.x = b2[kb];     f2.y = ok ? b2[kb + 1] : 0.f;
      f3.x = b3[kb];     f3.y = ok ? b3[kb + 1] : 0.f;
      acc0 = __builtin_amdgcn_wmma_f32_16x16x4_f32(false, a, false, f0, (short)0, acc0, false, false);
      acc1 = __builtin_amdgcn_wmma_f32_16x16x4_f32(false, a, false, f1, (short)0, acc1, false, false);
      acc2 = __builtin_amdgcn_wmma_f32_16x16x4_f32(false, a, false, f2, (short)0, acc2, false, false);
      acc3 = __builtin_amdgcn_wmma_f32_16x16x4_f32(false, a, false, f3, (short)0, acc3, false, false);
    }
  }

  // Epilogue: C/D layout (N = lane&15, M = v + 8*(lane>=16)) -> global store.
  int row0 = rowbase + 8 * hi;
  const v8f* accs[4] = { &acc0, &acc1, &acc2, &acc3 };
#pragma unroll
  for (int c = 0; c < 4; ++c) {
    int t = tbase + c * 16 + lo;
    if (t < Rn) {
      float rr = (OUT_ID == 2) ? root_role[t] : 0.f;
      v8f av = *accs[c];
#pragma unroll
      for (int v = 0; v < 8; ++v) {
        int row = row0 + v;
        float val = av[v];
        if (OUT_ID == 2) val += root_filler[row] * rr;
        dst[(size_t)row * Rn + t] = val;
      }
    }
  }
}

__global__ void __launch_bounds__(256)
dtm_gemm(const float* __restrict__ ws,
         const float* __restrict__ Dl, const float* __restrict__ Dr,
         const float* __restrict__ El, const float* __restrict__ Er,
         const float* __restrict__ root_filler,
         const float* __restrict__ root_role,
         float* __restrict__ out) {
  int grp  = blockIdx.x;            // 0..1535 : scalar output selector
  int wib  = threadIdx.x >> 5;      // wave in block (8 waves)
  int lane = threadIdx.x & 31;
  if (grp < 512) {
    gemm_path<0>(grp * 8 + wib, lane, ws, Dl, nullptr,
                 root_filler, root_role, out);
  } else if (grp < 1024) {
    gemm_path<1>((grp - 512) * 8 + wib, lane, ws + (size_t)ARGSZ, Dr, nullptr,
                 root_filler, root_role, out + (size_t)ARGSZ);
  } else {
    gemm_path<2>((grp - 1024) * 8 + wib, lane, ws + (size_t)2 * ARGSZ, El, Er,
                 root_filler, root_role, out + (size_t)2 * ARGSZ);
  }
}

// ---------------------------------------------------------------------------
extern "C" void kernel_launch(void* const* d_in, const int* in_sizes, int n_in,
                              void* d_out, int out_size, void* d_ws, size_t ws_size,
                              hipStream_t stream) {
  const float* x           = (const float*)d_in[0];
  const float* car_w       = (const float*)d_in[1];
  const float* cdr_w       = (const float*)d_in[2];
  const float* cons1_w     = (const float*)d_in[3];
  const float* cons2_w     = (const float*)d_in[4];
  const float* root_filler = (const float*)d_in[5];
  const float* Dl          = (const float*)d_in[6];
  const float* Dr          = (const float*)d_in[7];
  const float* El          = (const float*)d_in[8];
  const float* Er          = (const float*)d_in[9];
  const float* root_role   = (const float*)d_in[10];
  float* out = (float*)d_out;
  float* ws  = (float*)d_ws;

  // Stage 1: one pass over x, 4 weighted sums into workspace (67 MB).
  dtm_stage1<<<(ARGSZ + 255) / 256, 256, 0, stream>>>(
      x, car_w, cdr_w, cons1_w, cons2_w, ws);

  // Entropies + maxes (trivial).
  dtm_stats<<<1, 256, 0, stream>>>(car_w, cdr_w, cons1_w, cons2_w, out);

  // Stage 2: (512 + 512 + 512) blocks x 8 waves = 12288 waves total.
  dtm_gemm<<<1536, 256, 0, stream>>>(
      ws, Dl, Dr, El, Er, root_filler, root_role, out);
}